// DifferentiableProjectionLayer_4294967296186
// MI455X (gfx1250) — compile-verified
//
#include <hip/hip_runtime.h>

#define NLINK   4096
#define BLOCK   256
#define RPB     (BLOCK / 32)        // 8 rows per block: one wave32 per row
#define CHUNKS  (NLINK / 32 / 4)    // 32 float4 chunks per lane (128 floats/lane)
#define BUDGETF 100.0f
#define PMAXF   0.1f
#define NBISECT 60

typedef float v2f __attribute__((ext_vector_type(2)));
typedef float v8f __attribute__((ext_vector_type(8)));

// Swap values between the two 16-lane halves and add (VALU v_permlanex16, no LDS).
__device__ __forceinline__ float xhalf_swap_add(float t) {
#if defined(__has_builtin) && __has_builtin(__builtin_amdgcn_permlanex16)
    int o = __builtin_amdgcn_permlanex16(
        __builtin_bit_cast(int, t), __builtin_bit_cast(int, t),
        0x76543210u, 0xfedcba98u, false, false);
    return t + __builtin_bit_cast(float, o);
#else
    return t + __shfl_xor(t, 16, 32);
#endif
}

// Full-precision f32 wave(32) sum using V_WMMA_F32_16X16X4_F32.
// A: VGPR0 = per-lane value, VGPR1 = 0  => A[m][0]=s_m, A[m][2]=s_{m+16}
// B: all ones (4x16)                    => D[m][n] = s_m + s_{m+16}
// Sum of D's 8 VGPRs gives 16-lane-group half sums; one cross-half
// permute+add completes the 32-lane sum, broadcast to every lane.
__device__ __forceinline__ float wave_sum_wmma(float x) {
    v2f a; a[0] = x;    a[1] = 0.0f;
    v2f b; b[0] = 1.0f; b[1] = 1.0f;
    v8f c = {};
    v8f d = __builtin_amdgcn_wmma_f32_16x16x4_f32(
        /*neg_a=*/false, a, /*neg_b=*/false, b,
        /*c_mod=*/(short)0, c, /*reuse_a=*/false, /*reuse_b=*/false);
    float t = ((d[0] + d[1]) + (d[2] + d[3])) + ((d[4] + d[5]) + (d[6] + d[7]));
    return xhalf_swap_add(t);
}

__global__ __launch_bounds__(BLOCK)
void proj_wf_kernel(const float* __restrict__ raw, float* __restrict__ out, int rows) {
    const int lane = threadIdx.x & 31;
    const int wid  = threadIdx.x >> 5;
    const int row  = blockIdx.x * RPB + wid;
    if (row >= rows) return;   // wave-uniform guard

    const float* __restrict__ rp = raw + (size_t)row * NLINK;
    float*       __restrict__ op = out + (size_t)row * NLINK;

    // ---- load the full row slice into registers (32 x global_load_b128) ----
    float4 x[CHUNKS];
#pragma unroll
    for (int i = 0; i < CHUNKS; ++i) x[i] = ((const float4*)rp)[i * 32 + lane];

    // ---- local clip-sum / min / max (4 accumulators for ILP) ----
    float s0 = 0.f, s1 = 0.f, s2 = 0.f, s3 = 0.f;
    float lmin = x[0].x, lmax = x[0].x;
#pragma unroll
    for (int i = 0; i < CHUNKS; ++i) {
        s0 += __builtin_amdgcn_fmed3f(x[i].x, 0.0f, PMAXF);
        s1 += __builtin_amdgcn_fmed3f(x[i].y, 0.0f, PMAXF);
        s2 += __builtin_amdgcn_fmed3f(x[i].z, 0.0f, PMAXF);
        s3 += __builtin_amdgcn_fmed3f(x[i].w, 0.0f, PMAXF);
        lmin = fminf(lmin, fminf(fminf(x[i].x, x[i].y), fminf(x[i].z, x[i].w)));
        lmax = fmaxf(lmax, fmaxf(fmaxf(x[i].x, x[i].y), fmaxf(x[i].z, x[i].w)));
    }
    const float tot = wave_sum_wmma((s0 + s1) + (s2 + s3));
    float rmin = lmin, rmax = lmax;
#pragma unroll
    for (int o = 16; o > 0; o >>= 1) {
        rmin = fminf(rmin, __shfl_xor(rmin, o, 32));
        rmax = fmaxf(rmax, __shfl_xor(rmax, o, 32));
    }

    // ---- feasible: output plain clip (wave-uniform branch, EXEC stays full) ----
    if (tot <= BUDGETF) {
#pragma unroll
        for (int i = 0; i < CHUNKS; ++i) {
            // in-place: result overwrites x[i], no extra live registers
            x[i].x = __builtin_amdgcn_fmed3f(x[i].x, 0.0f, PMAXF);
            x[i].y = __builtin_amdgcn_fmed3f(x[i].y, 0.0f, PMAXF);
            x[i].z = __builtin_amdgcn_fmed3f(x[i].z, 0.0f, PMAXF);
            x[i].w = __builtin_amdgcn_fmed3f(x[i].w, 0.0f, PMAXF);
            ((float4*)op)[i * 32 + lane] = x[i];
        }
        return;
    }

    // ---- bisection on tau: g(tau) = sum(clip(x - tau, 0, PMAX)) ----
    float lo = rmin - PMAXF;   // g(lo) = n*PMAX >= BUDGET
    float hi = rmax;           // g(hi) = 0
#pragma unroll 1
    for (int it = 0; it < NBISECT; ++it) {
        const float mid = 0.5f * (lo + hi);
        float a0 = 0.f, a1 = 0.f, a2 = 0.f, a3 = 0.f;
#pragma unroll
        for (int i = 0; i < CHUNKS; ++i) {
            a0 += __builtin_amdgcn_fmed3f(x[i].x - mid, 0.0f, PMAXF);
            a1 += __builtin_amdgcn_fmed3f(x[i].y - mid, 0.0f, PMAXF);
            a2 += __builtin_amdgcn_fmed3f(x[i].z - mid, 0.0f, PMAXF);
            a3 += __builtin_amdgcn_fmed3f(x[i].w - mid, 0.0f, PMAXF);
        }
        const float g = wave_sum_wmma((a0 + a1) + (a2 + a3));
        // g identical in every lane -> uniform update, EXEC stays all-ones
        if (g > BUDGETF) lo = mid; else hi = mid;
    }
    const float tau0 = 0.5f * (lo + hi);

    // ---- Newton correction: tau = tau0 + (g0 - BUDGET) / n_active ----
    float lg = 0.f, lc = 0.f;
#pragma unroll
    for (int i = 0; i < CHUNKS; ++i) {
        const float dx0 = x[i].x - tau0, dx1 = x[i].y - tau0;
        const float dx2 = x[i].z - tau0, dx3 = x[i].w - tau0;
        lg += __builtin_amdgcn_fmed3f(dx0, 0.0f, PMAXF)
            + __builtin_amdgcn_fmed3f(dx1, 0.0f, PMAXF)
            + __builtin_amdgcn_fmed3f(dx2, 0.0f, PMAXF)
            + __builtin_amdgcn_fmed3f(dx3, 0.0f, PMAXF);
        lc += ((dx0 > 0.0f && dx0 < PMAXF) ? 1.0f : 0.0f)
            + ((dx1 > 0.0f && dx1 < PMAXF) ? 1.0f : 0.0f)
            + ((dx2 > 0.0f && dx2 < PMAXF) ? 1.0f : 0.0f)
            + ((dx3 > 0.0f && dx3 < PMAXF) ? 1.0f : 0.0f);
    }
    const float g0       = wave_sum_wmma(lg);
    const float n_active = fmaxf(wave_sum_wmma(lc), 1.0f);
    const float tau      = tau0 + (g0 - BUDGETF) / n_active;

    // ---- output p = clip(x - tau, 0, PMAX), in-place then b128 store ----
#pragma unroll
    for (int i = 0; i < CHUNKS; ++i) {
        x[i].x = __builtin_amdgcn_fmed3f(x[i].x - tau, 0.0f, PMAXF);
        x[i].y = __builtin_amdgcn_fmed3f(x[i].y - tau, 0.0f, PMAXF);
        x[i].z = __builtin_amdgcn_fmed3f(x[i].z - tau, 0.0f, PMAXF);
        x[i].w = __builtin_amdgcn_fmed3f(x[i].w - tau, 0.0f, PMAXF);
        ((float4*)op)[i * 32 + lane] = x[i];
    }
}

extern "C" void kernel_launch(void* const* d_in, const int* in_sizes, int n_in,
                              void* d_out, int out_size, void* d_ws, size_t ws_size,
                              hipStream_t stream) {
    (void)n_in; (void)out_size; (void)d_ws; (void)ws_size;
    const float* raw = (const float*)d_in[0];
    float*       out = (float*)d_out;
    const int rows = in_sizes[0] / NLINK;   // 8192
    const int grid = (rows + RPB - 1) / RPB;
    proj_wf_kernel<<<grid, BLOCK, 0, stream>>>(raw, out, rows);
}